// DeepseekV2Attention_8718783611696
// MI455X (gfx1250) — compile-verified
//
#include <hip/hip_runtime.h>
#include <hip/hip_bf16.h>

// ---- problem constants ----
#define NB      2
#define NS      2048
#define NHID    2048
#define NH      16
#define NQLORA  1536
#define NKVLORA 512
#define NDNOPE  128
#define NDROPE  64
#define NDV     128
#define NDQ     192
#define NM      (NB * NS)          // 4096 token rows
#define EPSV    1e-6f

typedef __attribute__((ext_vector_type(2))) float v2f;
typedef __attribute__((ext_vector_type(8))) float v8f;

// fp32 WMMA: D(16x16,f32) = A(16x4,f32) * B(4x16,f32) + C
__device__ __forceinline__ v8f wmma4(v2f a, v2f b, v8f c) {
  return __builtin_amdgcn_wmma_f32_16x16x4_f32(false, a, false, b, (short)0, c,
                                               false, false);
}

__device__ __forceinline__ v2f ld2(const float* __restrict__ base, int idx) {
  return *(const v2f*)(base + idx);   // uniform base + 32-bit lane offset -> GVS mode
}

// ---------------------------------------------------------------------------
// Generic fp32 GEMM: C[M,N] = A[M,K] @ B[K,N], row-major.
// One wave computes a 32(M) x 64(N) tile (2x4 grid of 16x16 WMMA tiles):
// every B fragment is reused by 2 A fragments -> 1.25 loads per WMMA.
// All weights (~61MB) fit in the 192MB L2, so direct global loads stream from
// L2 after first touch; a light global_prefetch_b8 hides the first-touch hop.
// Addressing: uniform SGPR base + 32-bit int lane offsets (GVS mode) to keep
// the address stream to one 32-bit VALU add per pointer per k-step.
// ---------------------------------------------------------------------------
__global__ void __launch_bounds__(256) gemm_f32_wmma(
    const float* __restrict__ A, const float* __restrict__ Bm,
    float* __restrict__ C, int M, int N, int K) {
  const int lane = threadIdx.x & 31;
  const int wave = threadIdx.x >> 5;
  const int n    = lane & 15;       // col within 16 (also A-fragment row)
  const int khi  = lane >> 4;       // K-half selector
  const int tilesN = N >> 6;
  int wt = blockIdx.x * (blockDim.x >> 5) + wave;
  if (wt >= (M >> 5) * tilesN) return;
  const int row0 = (wt / tilesN) << 5;   // 32-row tile
  const int col0 = (wt % tilesN) << 6;   // 64-col tile

  v8f acc[2][4];
#pragma unroll
  for (int i = 0; i < 2; ++i)
#pragma unroll
    for (int j = 0; j < 4; ++j) acc[i][j] = (v8f){};

  const int aoff0 = (row0 + n) * K + 2 * khi;   // 32-bit lane offsets
  const int aoff1 = aoff0 + 16 * K;
  const int boff  = (2 * khi) * N + col0 + n;
#pragma unroll 4
  for (int k = 0; k < K; k += 4) {
    v2f a0 = ld2(A, aoff0 + k);
    v2f a1 = ld2(A, aoff1 + k);
    const int bk = boff + k * N;
    __builtin_prefetch(Bm + bk + 64 * N, 0, 1);  // 16 k-steps ahead
    v2f b0 = { Bm[bk],      Bm[bk + N] };
    v2f b1 = { Bm[bk + 16], Bm[bk + N + 16] };
    v2f b2 = { Bm[bk + 32], Bm[bk + N + 32] };
    v2f b3 = { Bm[bk + 48], Bm[bk + N + 48] };
    acc[0][0] = wmma4(a0, b0, acc[0][0]);
    acc[0][1] = wmma4(a0, b1, acc[0][1]);
    acc[0][2] = wmma4(a0, b2, acc[0][2]);
    acc[0][3] = wmma4(a0, b3, acc[0][3]);
    acc[1][0] = wmma4(a1, b0, acc[1][0]);
    acc[1][1] = wmma4(a1, b1, acc[1][1]);
    acc[1][2] = wmma4(a1, b2, acc[1][2]);
    acc[1][3] = wmma4(a1, b3, acc[1][3]);
  }
#pragma unroll
  for (int i = 0; i < 2; ++i) {
    const int cbase = (row0 + 16 * i) * N + col0 + n;
#pragma unroll
    for (int v = 0; v < 8; ++v) {
      const int off = cbase + (v + 8 * khi) * N;
      C[off]      = acc[i][0][v];
      C[off + 16] = acc[i][1][v];
      C[off + 32] = acc[i][2][v];
      C[off + 48] = acc[i][3][v];
    }
  }
}

// ---------------------------------------------------------------------------
// RMSNorm over rows: y = x * rsqrt(mean(x^2)+eps) * w.  One 256-thread block
// (8 waves) per row; supports strided input (for the 576->512 ckv slice).
// ---------------------------------------------------------------------------
__global__ void __launch_bounds__(256) rmsnorm_rows(
    const float* __restrict__ x, const float* __restrict__ w,
    float* __restrict__ y, int len, int xstride, int ystride) {
  const int row = blockIdx.x;
  const float* xr = x + (size_t)row * xstride;
  float* yr = y + (size_t)row * ystride;
  float ss = 0.f;
  for (int i = threadIdx.x; i < len; i += blockDim.x) {
    float v = xr[i];
    ss += v * v;
  }
#pragma unroll
  for (int m = 16; m; m >>= 1) ss += __shfl_xor(ss, m, 32);
  __shared__ float red[8];
  const int wave = threadIdx.x >> 5, lane = threadIdx.x & 31;
  if (lane == 0) red[wave] = ss;
  __syncthreads();
  float total = 0.f;
  const int nw = blockDim.x >> 5;
  for (int i = 0; i < nw; ++i) total += red[i];
  const float r = rsqrtf(total / (float)len + EPSV);
  for (int i = threadIdx.x; i < len; i += blockDim.x) yr[i] = xr[i] * r * w[i];
}

// ---------------------------------------------------------------------------
// RoPE (DeepSeek interleaved->split permutation), in place, one thread per
// 64-dim vector. Output kept in permuted order (consistent for q and k, so
// scores are unchanged).
// ---------------------------------------------------------------------------
__device__ __forceinline__ void rope_apply64(float* v, int t) {
  float x[64];
#pragma unroll
  for (int i = 0; i < 64; ++i) x[i] = v[i];
#pragma unroll
  for (int i = 0; i < 32; ++i) {
    // inv_freq = 10000^(-2i/64)
    float invf = __expf(-(2.0f * (float)i / 64.0f) * 9.210340371976184f);
    float ang = (float)t * invf;
    float sn, cs;
    __sincosf(ang, &sn, &cs);
    float xe = x[2 * i], xo = x[2 * i + 1];
    v[i]      = xe * cs - xo * sn;
    v[32 + i] = xo * cs + xe * sn;
  }
}

__global__ void __launch_bounds__(256) rope_kernel(float* __restrict__ q,
                                                   float* __restrict__ ckv) {
  const int idx = blockIdx.x * blockDim.x + threadIdx.x;
  const int nq = NB * NS * NH;   // q_pe vectors
  const int nk = NB * NS;        // k_pe vectors
  if (idx < nq) {
    int s = (idx / NH) % NS;
    rope_apply64(q + (size_t)idx * NDQ + NDNOPE, s);
  } else if (idx < nq + nk) {
    int j = idx - nq;
    int s = j % NS;
    rope_apply64(ckv + (size_t)j * (NKVLORA + NDROPE) + NKVLORA, s);
  }
}

// ---------------------------------------------------------------------------
// Flash-style causal MLA attention, fp32 WMMA for QK^T and P*V.
// One wave per (b, h, 16-row q tile). P staged through wave-private LDS to
// convert C-layout -> A-fragment layout.  32-bit lane offsets off uniform
// bases keep addressing in GVS mode.
//   q   : [B,S,H,192]   (q_nope | roped q_pe)
//   kv  : [B,S,H,256]   (k_nope | v)
//   ckv : [B,S,576]     (roped k_pe at +512, shared by heads)
//   o   : [B,S,H,128]
// ---------------------------------------------------------------------------
__global__ void __launch_bounds__(256) mla_attn_wmma(
    const float* __restrict__ q, const float* __restrict__ kv,
    const float* __restrict__ ckv, float* __restrict__ o) {
  __shared__ float pbuf[8][16 * 17];   // per-wave P tile, pitch 17 (no bank conflicts)
  const int lane = threadIdx.x & 31;
  const int wave = threadIdx.x >> 5;
  const int n    = lane & 15;
  const int khi  = lane >> 4;

  const int qtiles = NS / 16;
  int tile = blockIdx.x * (blockDim.x >> 5) + wave;
  const int qt = tile % qtiles;
  const int bh = tile / qtiles;
  const int h  = bh % NH;
  const int b  = bh / NH;
  const int qt0 = qt * 16;

  // A-fragment row for both QK^T and P*V is (lane & 15)
  const int qoff = ((b * NS + qt0 + n) * NH + h) * NDQ;   // max ~12.6M, fits int

  v8f oacc[8];
  float mi[8], li[8];
#pragma unroll
  for (int c = 0; c < 8; ++c) oacc[c] = (v8f){};
#pragma unroll
  for (int v = 0; v < 8; ++v) { mi[v] = -3.0e38f; li[v] = 0.f; }

  const float scale = 0.07216878364870323f;  // 1/sqrt(192)

  for (int kt0 = 0; kt0 <= qt0; kt0 += 16) {
    // ---- S = Q . K^T (this 16x16 tile) ----
    v8f sacc = (v8f){};
    const int koff = ((b * NS + kt0 + n) * NH + h) * (NDNOPE + NDV);
#pragma unroll 8
    for (int k = 0; k < NDNOPE; k += 4) {
      v2f a  = ld2(q,  qoff + k + 2 * khi);
      v2f bf = ld2(kv, koff + k + 2 * khi);
      sacc = wmma4(a, bf, sacc);
    }
    const int peoff = (b * NS + kt0 + n) * (NKVLORA + NDROPE) + NKVLORA;
#pragma unroll 8
    for (int k = 0; k < NDROPE; k += 4) {
      v2f a  = ld2(q,   qoff + NDNOPE + k + 2 * khi);
      v2f bf = ld2(ckv, peoff + k + 2 * khi);
      sacc = wmma4(a, bf, sacc);
    }

    // ---- online softmax (rows r = v + 8*khi live in the same 16-lane half) ----
    const bool diag = (kt0 == qt0);
    float alpha[8];
#pragma unroll
    for (int v = 0; v < 8; ++v) {
      const int r = v + 8 * khi;
      float sv = sacc[v] * scale;
      if (diag && n > r) sv = -3.0e38f;           // causal mask
      float rm = sv;
      rm = fmaxf(rm, __shfl_xor(rm, 1, 32));
      rm = fmaxf(rm, __shfl_xor(rm, 2, 32));
      rm = fmaxf(rm, __shfl_xor(rm, 4, 32));
      rm = fmaxf(rm, __shfl_xor(rm, 8, 32));
      float mn = fmaxf(mi[v], rm);
      float p = __expf(sv - mn);
      float rs = p;
      rs += __shfl_xor(rs, 1, 32);
      rs += __shfl_xor(rs, 2, 32);
      rs += __shfl_xor(rs, 4, 32);
      rs += __shfl_xor(rs, 8, 32);
      alpha[v] = __expf(mi[v] - mn);
      li[v] = li[v] * alpha[v] + rs;
      mi[v] = mn;
      pbuf[wave][r * 17 + n] = p;                 // wave-private, DS in-order
    }
#pragma unroll
    for (int c = 0; c < 8; ++c)
#pragma unroll
      for (int v = 0; v < 8; ++v) oacc[c][v] *= alpha[v];

    // ---- O += P . V  (K = 16 keys, 4 WMMA k-steps, 8 v-dim chunks) ----
#pragma unroll
    for (int ks = 0; ks < 4; ++ks) {
      v2f a;
      a[0] = pbuf[wave][n * 17 + ks * 4 + 2 * khi];
      a[1] = pbuf[wave][n * 17 + ks * 4 + 2 * khi + 1];
      const int key = kt0 + ks * 4 + 2 * khi;
      const int voff = ((b * NS + key) * NH + h) * (NDNOPE + NDV) + NDNOPE;
      const int vstride = NH * (NDNOPE + NDV);
#pragma unroll
      for (int c = 0; c < 8; ++c) {
        v2f bf = { kv[voff + c * 16 + n], kv[voff + vstride + c * 16 + n] };
        oacc[c] = wmma4(a, bf, oacc[c]);
      }
    }
  }

  // ---- epilogue: normalize and store ----
#pragma unroll
  for (int v = 0; v < 8; ++v) {
    const int r = v + 8 * khi;
    const float inv = 1.0f / li[v];
    const int ooff = ((b * NS + qt0 + r) * NH + h) * NDV;
#pragma unroll
    for (int c = 0; c < 8; ++c) o[ooff + c * 16 + n] = oacc[c][v] * inv;
  }
}

// ---------------------------------------------------------------------------
// Host launcher
// ---------------------------------------------------------------------------
static inline int gemm_grid(int M, int N) { return (M / 32) * (N / 64) / 8; }

extern "C" void kernel_launch(void* const* d_in, const int* in_sizes, int n_in,
                              void* d_out, int out_size, void* d_ws, size_t ws_size,
                              hipStream_t stream) {
  const float* x       = (const float*)d_in[0];  // [B,S,2048]
  const float* w_dq    = (const float*)d_in[1];  // [2048,1536]
  const float* q_ln_w  = (const float*)d_in[2];  // [1536]
  const float* w_uq    = (const float*)d_in[3];  // [1536,3072]
  const float* kv_a_w  = (const float*)d_in[4];  // [2048,576]
  const float* kv_ln_w = (const float*)d_in[5];  // [512]
  const float* kv_b_w  = (const float*)d_in[6];  // [512,4096]
  const float* o_w     = (const float*)d_in[7];  // [2048,2048]
  float* out = (float*)d_out;

  // workspace partition (floats)
  float* ws     = (float*)d_ws;
  float* q_lora = ws;                                       // 4096*1536
  float* qbuf   = q_lora + (size_t)NM * NQLORA;             // 4096*3072
  float* ckv    = qbuf   + (size_t)NM * (NH * NDQ);         // 4096*576
  float* ckv_ln = ckv    + (size_t)NM * (NKVLORA + NDROPE); // 4096*512
  float* kvbuf  = ckv_ln + (size_t)NM * NKVLORA;            // 4096*4096
  float* attn   = kvbuf  + (size_t)NM * (NH * (NDNOPE + NDV)); // 4096*2048

  // 1) q_lora = x @ w_dq
  gemm_f32_wmma<<<gemm_grid(NM, NQLORA), 256, 0, stream>>>(x, w_dq, q_lora,
                                                           NM, NQLORA, NHID);
  // 2) rmsnorm(q_lora) in place
  rmsnorm_rows<<<NM, 256, 0, stream>>>(q_lora, q_ln_w, q_lora,
                                       NQLORA, NQLORA, NQLORA);
  // 3) q = q_lora @ w_uq  -> [B,S,H,192]
  gemm_f32_wmma<<<gemm_grid(NM, NH * NDQ), 256, 0, stream>>>(q_lora, w_uq, qbuf,
                                                             NM, NH * NDQ, NQLORA);
  // 4) ckv_pe = x @ kv_a_w -> [B,S,576]
  gemm_f32_wmma<<<gemm_grid(NM, NKVLORA + NDROPE), 256, 0, stream>>>(
      x, kv_a_w, ckv, NM, NKVLORA + NDROPE, NHID);
  // 5) ckv_ln = rmsnorm(ckv[:, :512])
  rmsnorm_rows<<<NM, 256, 0, stream>>>(ckv, kv_ln_w, ckv_ln,
                                       NKVLORA, NKVLORA + NDROPE, NKVLORA);
  // 6) kv = ckv_ln @ kv_b_w -> [B,S,H,256]
  gemm_f32_wmma<<<gemm_grid(NM, NH * (NDNOPE + NDV)), 256, 0, stream>>>(
      ckv_ln, kv_b_w, kvbuf, NM, NH * (NDNOPE + NDV), NKVLORA);
  // 7) RoPE on q_pe and k_pe (in place)
  {
    const int tot = NB * NS * NH + NB * NS;
    rope_kernel<<<(tot + 255) / 256, 256, 0, stream>>>(qbuf, ckv);
  }
  // 8) flash attention -> attn [B,S,H*128]
  mla_attn_wmma<<<NB * NH * (NS / 16) / 8, 256, 0, stream>>>(qbuf, kvbuf, ckv, attn);
  // 9) out = attn @ o_w
  gemm_f32_wmma<<<gemm_grid(NM, NHID), 256, 0, stream>>>(attn, o_w, out,
                                                         NM, NHID, NH * NDV);
}